// MentionLabeler_57208964382800
// MI455X (gfx1250) — compile-verified
//
#include <hip/hip_runtime.h>

typedef __attribute__((ext_vector_type(2))) float v2f;
typedef __attribute__((ext_vector_type(8))) float v8f;

#define BB   4
#define NN   512
#define HH   768
#define HID  256
#define MTOT (BB * NN)   // 2048 rows

// ---------------------------------------------------------------------------
// Kernel 1: dual GEMM via V_WMMA_F32_16X16X4_F32.
//   aP[m,n] = emb[m,:] @ Wa[:,n]
//   cP[m,n] = emb[m,:] @ Wb[:,n] + b1[n]
// One wave32 computes one 16x16 tile of BOTH outputs (A-fragment reuse).
// A-frag (16x4 f32):  lane l, VGPR v -> A[m0 + l%16][k0 + v + 2*(l/16)]
// B-frag (4x16 f32):  lane l, VGPR v -> B[k0 + v + 2*(l/16)][n0 + l%16]
// C/D    (16x16 f32): lane l, VGPR r -> D[m0 + r + 8*(l/16)][n0 + l%16]
// ---------------------------------------------------------------------------
__global__ __launch_bounds__(128) void gemm_dual_wmma(
    const float* __restrict__ emb, const float* __restrict__ W1,
    const float* __restrict__ b1,
    float* __restrict__ aP, float* __restrict__ cP)
{
  const int lane  = threadIdx.x & 31;
  const int wave  = blockIdx.x * (blockDim.x >> 5) + (threadIdx.x >> 5);
  const int mTile = wave >> 4;          // 0..127
  const int nTile = wave & 15;          // 0..15
  const int m0 = mTile << 4;
  const int n0 = nTile << 4;
  const int lm = lane & 15;
  const int h  = lane >> 4;

  const float* __restrict__ Wa = W1;              // rows 0..767
  const float* __restrict__ Wb = W1 + HH * HID;   // rows 768..1535
  const float* __restrict__ aRow = emb + (size_t)(m0 + lm) * HH;

  v8f accA = {};
  v8f accC = {};

  for (int k0 = 0; k0 < HH; k0 += 4) {
    const int ka = k0 + 2 * h;
    v2f afrag, bA, bC;
    afrag.x = aRow[ka];
    afrag.y = aRow[ka + 1];
    bA.x = Wa[(size_t)ka * HID + n0 + lm];
    bA.y = Wa[(size_t)(ka + 1) * HID + n0 + lm];
    bC.x = Wb[(size_t)ka * HID + n0 + lm];
    bC.y = Wb[(size_t)(ka + 1) * HID + n0 + lm];
    accA = __builtin_amdgcn_wmma_f32_16x16x4_f32(false, afrag, false, bA,
                                                 (short)0, accA, false, false);
    accC = __builtin_amdgcn_wmma_f32_16x16x4_f32(false, afrag, false, bC,
                                                 (short)0, accC, false, false);
  }

  const int n = n0 + lm;
  const float bias = b1[n];
#pragma unroll
  for (int r = 0; r < 8; ++r) {
    const int m = m0 + r + 8 * h;
    aP[(size_t)m * HID + n] = accA[r];
    cP[(size_t)m * HID + n] = accC[r] + bias;
  }
}

// ---------------------------------------------------------------------------
// Kernel 2: fused pairwise relu-dot (the 268M-element hot loop, VALU-bound).
//   out[b,i,j] = b2 + sum_d relu(aP[b,i,d] + cP[b,j,d]) * w2[d]
// 64x64 output tile per block (256 threads), 4x4 micro-tile per thread.
// LDS rows padded to 65 floats -> conflict-free strided column reads.
// ---------------------------------------------------------------------------
#define TILE 64
#define DCH  64
#define LSTR 65

__global__ __launch_bounds__(256) void pairwise_relu_dot(
    const float* __restrict__ aP, const float* __restrict__ cP,
    const float* __restrict__ w2, const float* __restrict__ b2,
    float* __restrict__ out)
{
  __shared__ float As[TILE * LSTR];
  __shared__ float Cs[TILE * LSTR];
  __shared__ float Ws[HID];

  const int t  = threadIdx.x;
  const int tx = t & 15;        // j direction
  const int ty = t >> 4;        // i direction
  const int bb = blockIdx.z;
  const int i0 = blockIdx.y * TILE;
  const int j0 = blockIdx.x * TILE;

  const float* __restrict__ aBase = aP + (size_t)(bb * NN + i0) * HID;
  const float* __restrict__ cBase = cP + (size_t)(bb * NN + j0) * HID;
  Ws[t] = w2[t];                // blockDim == HID == 256

  float acc[16];
#pragma unroll
  for (int q = 0; q < 16; ++q) acc[q] = 0.f;

  for (int dc = 0; dc < HID; dc += DCH) {
    __syncthreads();            // also publishes Ws on first pass
#pragma unroll
    for (int qq = 0; qq < 4; ++qq) {
      const int ri = qq * 16 + ty;
      const float4 av = *(const float4*)(aBase + (size_t)ri * HID + dc + 4 * tx);
      const float4 cv = *(const float4*)(cBase + (size_t)ri * HID + dc + 4 * tx);
      float* as = &As[ri * LSTR + 4 * tx];
      as[0] = av.x; as[1] = av.y; as[2] = av.z; as[3] = av.w;
      float* cs = &Cs[ri * LSTR + 4 * tx];
      cs[0] = cv.x; cs[1] = cv.y; cs[2] = cv.z; cs[3] = cv.w;
    }
    __syncthreads();

#pragma unroll 4
    for (int dd = 0; dd < DCH; ++dd) {
      const float w = Ws[dc + dd];
      float av[4], cv[4];
#pragma unroll
      for (int r = 0; r < 4; ++r) av[r] = As[(4 * ty + r) * LSTR + dd];
#pragma unroll
      for (int s = 0; s < 4; ++s) cv[s] = Cs[(4 * tx + s) * LSTR + dd];
#pragma unroll
      for (int r = 0; r < 4; ++r)
#pragma unroll
        for (int s = 0; s < 4; ++s)
          acc[r * 4 + s] = fmaf(fmaxf(av[r] + cv[s], 0.f), w, acc[r * 4 + s]);
    }
  }

  const float bias = b2[0];
#pragma unroll
  for (int r = 0; r < 4; ++r) {
    float4 v;
    v.x = acc[r * 4 + 0] + bias;
    v.y = acc[r * 4 + 1] + bias;
    v.z = acc[r * 4 + 2] + bias;
    v.w = acc[r * 4 + 3] + bias;
    float* o = out + ((size_t)(bb * NN + i0 + 4 * ty + r)) * NN + j0 + 4 * tx;
    *(float4*)o = v;
  }
}

// ---------------------------------------------------------------------------
extern "C" void kernel_launch(void* const* d_in, const int* in_sizes, int n_in,
                              void* d_out, int out_size, void* d_ws, size_t ws_size,
                              hipStream_t stream) {
  const float* emb = (const float*)d_in[0];   // (4,512,768)
  const float* W1  = (const float*)d_in[1];   // (1536,256)
  const float* b1  = (const float*)d_in[2];   // (256,)
  const float* W2  = (const float*)d_in[3];   // (256,1)
  const float* b2  = (const float*)d_in[4];   // (1,)
  float* out = (float*)d_out;                 // (4,512,512)

  float* aP = (float*)d_ws;                   // 2048*256 floats
  float* cP = aP + (size_t)MTOT * HID;        // 2048*256 floats (c + b1 folded)

  // 2048 wave-tiles (128 m-tiles x 16 n-tiles), 4 waves per block
  gemm_dual_wmma<<<512, 128, 0, stream>>>(emb, W1, b1, aP, cP);

  // 8x8 j/i tiles x 4 batches
  pairwise_relu_dot<<<dim3(8, 8, 4), 256, 0, stream>>>(aP, cP, W2, b2, out);
}